// Baseline_Counter_43980465111187
// MI455X (gfx1250) — compile-verified
//
#include <hip/hip_runtime.h>

typedef __attribute__((ext_vector_type(2))) float v2f;
typedef __attribute__((ext_vector_type(8))) float v8f;

// D = A(16x4, all ones) x B(4x16, 64 data values) + C  -> every row of C holds
// the 16 column sums; sum of c[0] across the wave = 2 * (sum of the 64 inputs).
static __device__ __forceinline__ v8f wmma_sum_acc(v2f ones, v2f b, v8f c) {
  return __builtin_amdgcn_wmma_f32_16x16x4_f32(
      /*neg_a=*/false, ones, /*neg_b=*/false, b,
      /*c_mod=*/(short)0, c, /*reuse_a=*/false, /*reuse_b=*/false);
}

// ---------------- Kernel 1: per-patch sums via f32 WMMA ----------------
// One wave32 per patch. perr = sum (out-gt)^2 over patch, fsum = sum gt.
__global__ __launch_bounds__(256)
void patch_sums_kernel(const float* __restrict__ out,
                       const float* __restrict__ gt,
                       float* __restrict__ perr,
                       float* __restrict__ fsum,
                       int S, int k, int lk, int nchunks) {
  const int wave = (int)((blockIdx.x * 256u + threadIdx.x) >> 5);
  const int lane = (int)(threadIdx.x & 31u);
  const int b = wave >> 8;          // 256 patches per batch
  const int p = wave & 255;
  const int pY = p >> 4, pX = p & 15;
  const size_t base = (size_t)b * (size_t)S * (size_t)S
                    + (size_t)(pY * k) * (size_t)S + (size_t)(pX * k);
  const float* ob = out + base;
  const float* gb = gt + base;

  const v2f ones = {1.0f, 1.0f};
  v8f accE = {0.f,0.f,0.f,0.f,0.f,0.f,0.f,0.f};
  v8f accG = {0.f,0.f,0.f,0.f,0.f,0.f,0.f,0.f};

  const int e_lane = lane << 1;     // 2 consecutive elements per lane
  #pragma unroll 4
  for (int c = 0; c < nchunks; ++c) {
    const int e0  = (c << 6) + e_lane;
    const int row = e0 >> lk;       // k even => pair stays within one row
    const int col = e0 & (k - 1);
    const size_t off = (size_t)row * (size_t)S + (size_t)col;
    const float2 o2 = *reinterpret_cast<const float2*>(ob + off);
    const float2 g2 = *reinterpret_cast<const float2*>(gb + off);
    const float d0 = o2.x - g2.x, d1 = o2.y - g2.y;
    v2f be; be.x = d0 * d0; be.y = d1 * d1;
    v2f bg; bg.x = g2.x;    bg.y = g2.y;
    accE = wmma_sum_acc(ones, be, accE);
    accG = wmma_sum_acc(ones, bg, accG);
  }

  float se = accE[0];
  float sg = accG[0];
  #pragma unroll
  for (int m = 16; m >= 1; m >>= 1) {
    se += __shfl_xor(se, m, 32);
    sg += __shfl_xor(sg, m, 32);
  }
  if (lane == 0) {
    perr[wave] = se * 0.5f;   // lanes 16..31 duplicate the column sums
    fsum[wave] = sg * 0.5f;
  }
}

// ------------- Kernel 2: score, argmin (first-min), masked loss -------------
__global__ __launch_bounds__(256)
void score_loss_kernel(const float* __restrict__ perr,  // [4][2048]
                       const float* __restrict__ fsum,  // [4][2048]
                       float* __restrict__ d_out,
                       int* __restrict__ moe_ws) {
  float* loss_out  = d_out;
  float* moe_out   = d_out + 1 + 8u * 1024u * 1024u;
  float* score_out = moe_out + 2048;
  __shared__ float red[256];
  const int tid = (int)threadIdx.x;
  const float inv_numel[4] = {1.f/8388608.f, 1.f/2097152.f, 1.f/524288.f, 1.f/131072.f};
  float lsum = 0.f;
  for (int t = tid; t < 2048; t += 256) {
    const int b = t >> 8, p = t & 255;
    float pe[4];
    int best = 0; float bestv = 0.f;
    #pragma unroll
    for (int i = 0; i < 4; ++i) {
      const int kk = 64 >> i;
      const float e = perr[i * 2048 + t];
      const float f = fsum[i * 2048 + t];
      const float sc = e / (float)(kk * kk) + e / (f + 1e-10f);
      pe[i] = e;
      score_out[b * 1024 + i * 256 + p] = sc;
      if (i == 0 || sc < bestv) { bestv = sc; best = i; }   // first minimum
    }
    moe_out[t] = (float)best;
    moe_ws[t]  = best;
    #pragma unroll
    for (int i = 0; i < 4; ++i)
      if (best >= i) lsum += pe[i] * inv_numel[i];  // level i counts moe>=i
  }
  red[tid] = lsum;
  __syncthreads();
  for (int s = 128; s > 0; s >>= 1) {
    if (tid < s) red[tid] += red[tid + s];
    __syncthreads();
  }
  if (tid == 0) loss_out[0] = red[0];
}

// ------------- Kernel 3: assemble label_patch (B,1,1024,1024) -------------
__global__ __launch_bounds__(256)
void label_patch_kernel(const float* __restrict__ gt0, const float* __restrict__ gt1,
                        const float* __restrict__ gt2, const float* __restrict__ gt3,
                        const int* __restrict__ moe,
                        float* __restrict__ label /* = d_out + 1 (4B aligned) */) {
  const size_t tid = (size_t)blockIdx.x * 256 + threadIdx.x;  // 2,097,152 total
  const int b   = (int)(tid >> 18);         // 262144 quads per batch image
  const int rem = (int)(tid & 262143u);
  const int y   = rem >> 8;                 // 256 quads per row
  const int xg  = (rem & 255) << 2;         // base x of 4-wide quad
  const int pY = y >> 6, pX = xg >> 6;
  const int m = moe[b * 256 + (pY << 4) + pX];

  float v[4];
  if (m == 0) {
    const float4 g = *reinterpret_cast<const float4*>(
        gt0 + ((size_t)b << 20) + ((size_t)y << 10) + (size_t)xg);
    v[0] = g.x; v[1] = g.y; v[2] = g.z; v[3] = g.w;
  } else {
    const float* gts = (m == 1) ? gt1 : (m == 2) ? gt2 : gt3;
    const int S = 1024 >> m;
    const int k = 64 >> m;
    const int pad = (64 - k) >> 1;
    const int py = y & 63;
    const bool rowin = (py >= pad) && (py < pad + k);
    if (rowin) {
      const int gy = pY * k + (py - pad);
      const long rowbase = (long)b * S * S + (long)gy * S + (long)pX * k - pad;
      const int pxb = xg & 63;
      #pragma unroll
      for (int j = 0; j < 4; ++j) {
        const int px = pxb + j;
        v[j] = (px >= pad && px < pad + k) ? gts[rowbase + px] : 0.2f;
      }
    } else {
      v[0] = v[1] = v[2] = v[3] = 0.2f;
    }
  }
  float* o = label + ((size_t)b << 20) + ((size_t)y << 10) + (size_t)xg;
  o[0] = v[0]; o[1] = v[1]; o[2] = v[2]; o[3] = v[3];
}

extern "C" void kernel_launch(void* const* d_in, const int* in_sizes, int n_in,
                              void* d_out, int out_size, void* d_ws, size_t ws_size,
                              hipStream_t stream) {
  // setup_inputs() dict order: out0, gt0, out1, gt1, out2, gt2, out3, gt3
  const float* outs[4] = {(const float*)d_in[0], (const float*)d_in[2],
                          (const float*)d_in[4], (const float*)d_in[6]};
  const float* gts[4]  = {(const float*)d_in[1], (const float*)d_in[3],
                          (const float*)d_in[5], (const float*)d_in[7]};
  float* dout = (float*)d_out;
  float* perr = (float*)d_ws;              // [4][2048]
  float* fsum = perr + 4 * 2048;           // [4][2048]
  int*   moe  = (int*)(fsum + 4 * 2048);   // [2048]

  for (int i = 0; i < 4; ++i) {
    const int S = 1024 >> i, k = 64 >> i, lk = 6 - i;
    const int nchunks = (k * k) >> 6;      // 64-element chunks per patch
    // 256 blocks * 8 waves = 2048 waves = one wave per (batch,patch)
    patch_sums_kernel<<<256, 256, 0, stream>>>(outs[i], gts[i],
                                               perr + i * 2048, fsum + i * 2048,
                                               S, k, lk, nchunks);
  }
  score_loss_kernel<<<1, 256, 0, stream>>>(perr, fsum, dout, moe);
  label_patch_kernel<<<8192, 256, 0, stream>>>(gts[0], gts[1], gts[2], gts[3],
                                               moe, dout + 1);
}